// SplitLayer_17368847745440
// MI455X (gfx1250) — compile-verified
//
#include <hip/hip_runtime.h>

#define Hdim 128
#define NN 100000
#define NE 400000
#define NI 800000
#define BN_EPS 1e-5f

typedef float v2f __attribute__((ext_vector_type(2)));
typedef float v8f __attribute__((ext_vector_type(8)));

__device__ __forceinline__ v8f wmma4(v2f a, v2f b, v8f c) {
  // D = A(16x4) * B(4x16) + C(16x16), fp32 WMMA, wave32
  return __builtin_amdgcn_wmma_f32_16x16x4_f32(false, a, false, b, (short)0, c,
                                               false, false);
}

#define AS_STRIDE 132  // 128 + 4 pad; 132*4 bytes is 16B-multiple -> float4 ok

// out[M x 128] = transform(A) @ W (W: 128x128 row-major, L2/L0 resident)
// MODE 0: A = A0
// MODE 1: A = (1 + *epsp) * A0 + A1          (residual + aggregation fuse)
// MODE 2: A = relu(A0 * bnscale[c] + bnshift[c])   (BN+ReLU fuse)
// STATS: accumulate per-column sum/sumsq of raw output for next BatchNorm.
template <int MODE, int STATS>
__global__ __launch_bounds__(256) void gemm128(
    const float* __restrict__ A0, int lda0, const float* __restrict__ A1,
    int lda1, const float* __restrict__ epsp, const float* __restrict__ bnscale,
    const float* __restrict__ bnshift, const float* __restrict__ W,
    float* __restrict__ out, int M, float* __restrict__ sumG,
    float* __restrict__ sqG) {
  __shared__ __align__(16) float As[64 * AS_STRIDE];
  __shared__ float s_sum[128];
  __shared__ float s_sq[128];

  const int tid = threadIdx.x;
  const int lane = tid & 31;
  const int wv = tid >> 5;
  const int row0 = blockIdx.x * 64;

  if (STATS && tid < 128) {
    s_sum[tid] = 0.f;
    s_sq[tid] = 0.f;
  }

  float alpha = 0.f;
  if (MODE == 1) alpha = 1.0f + epsp[0];

  // Cooperative, coalesced staging of the 64x128 A tile (with fused transform).
  for (int i = tid; i < 64 * 32; i += 256) {
    const int r = i >> 5;
    const int c = (i & 31) << 2;
    const int row = row0 + r;
    float4 v;
    if (row < M) {
      float4 a = *(const float4*)(A0 + (long long)row * lda0 + c);
      if (MODE == 0) {
        v = a;
      } else if (MODE == 1) {
        float4 b = *(const float4*)(A1 + (long long)row * lda1 + c);
        v.x = fmaf(alpha, a.x, b.x);
        v.y = fmaf(alpha, a.y, b.y);
        v.z = fmaf(alpha, a.z, b.z);
        v.w = fmaf(alpha, a.w, b.w);
      } else {
        float4 sc = *(const float4*)(bnscale + c);
        float4 sh = *(const float4*)(bnshift + c);
        v.x = fmaxf(fmaf(a.x, sc.x, sh.x), 0.f);
        v.y = fmaxf(fmaf(a.y, sc.y, sh.y), 0.f);
        v.z = fmaxf(fmaf(a.z, sc.z, sh.z), 0.f);
        v.w = fmaxf(fmaf(a.w, sc.w, sh.w), 0.f);
      }
    } else {
      v = make_float4(0.f, 0.f, 0.f, 0.f);  // zero rows: inert for stats too
    }
    *(float4*)(As + r * AS_STRIDE + c) = v;
  }
  __syncthreads();

  // Wave w: rows [16*(w/2), +16), cols [64*(w%2), +64)  -> 4 WMMA tiles.
  const int wm = (wv >> 1) << 4;
  const int wn = (wv & 1) << 6;
  const int l15 = lane & 15;
  const int kap = (lane >> 4) << 1;  // K sub-offset per ISA A/B fragment layout

  v8f acc[4];
#pragma unroll
  for (int t = 0; t < 4; ++t)
#pragma unroll
    for (int r = 0; r < 8; ++r) acc[t][r] = 0.f;

  const float* Ap = As + (wm + l15) * AS_STRIDE + kap;
  const float* Wp = W + wn + l15;
#pragma unroll 4
  for (int k0 = 0; k0 < 128; k0 += 4) {
    v2f a = *(const v2f*)(Ap + k0);  // ds_load_b64: A frag (K=kap, kap+1)
    const float* w0 = Wp + (k0 + kap) * 128;
#pragma unroll
    for (int t = 0; t < 4; ++t) {
      v2f b;
      b.x = w0[t * 16];        // W[k][n]   (half-wave contiguous, L0 hit)
      b.y = w0[t * 16 + 128];  // W[k+1][n]
      acc[t] = wmma4(a, b, acc[t]);
    }
  }

  // C/D layout: VGPR r -> row wm+r (lanes 0-15) / wm+8+r (lanes 16-31), col=n0+l15
  const int rbase = row0 + wm + ((lane >> 4) << 3);
#pragma unroll
  for (int t = 0; t < 4; ++t) {
    const int col = wn + (t << 4) + l15;
    float s = 0.f, s2 = 0.f;
#pragma unroll
    for (int r = 0; r < 8; ++r) {
      const float v = acc[t][r];
      const int row = rbase + r;
      if (row < M) out[(long long)row * Hdim + col] = v;
      if (STATS) {
        s += v;
        s2 += v * v;
      }
    }
    if (STATS) {
      atomicAdd(&s_sum[col], s);  // ds_add_f32
      atomicAdd(&s_sq[col], s2);
    }
  }
  if (STATS) {
    __syncthreads();
    if (tid < 128)
      unsafeAtomicAdd(&sumG[tid], s_sum[tid]);
    else
      unsafeAtomicAdd(&sqG[tid - 128], s_sq[tid - 128]);
  }
}

// Fold BN stats into per-column scale/shift: y = relu(scale*x + shift)
__global__ void bn_finalize(const float* __restrict__ sum,
                            const float* __restrict__ sq,
                            const float* __restrict__ g,
                            const float* __restrict__ b, float invM,
                            float* __restrict__ scale,
                            float* __restrict__ shift) {
  const int c = threadIdx.x;
  const float m = sum[c] * invM;
  const float var = sq[c] * invM - m * m;
  const float sc = g[c] * rsqrtf(var + BN_EPS);
  scale[c] = sc;
  shift[c] = b[c] - m * sc;
}

// In-place BN+ReLU epilogue on an M x 128 buffer (final layer of each tower).
__global__ __launch_bounds__(256) void bn_apply(float* __restrict__ data,
                                                const float* __restrict__ scale,
                                                const float* __restrict__ shift,
                                                long long n4) {
  const long long i = (long long)blockIdx.x * 256 + threadIdx.x;
  if (i >= n4) return;
  const int c = (int)(i & 31) << 2;
  float4 v = *(float4*)(data + i * 4);
  float4 sc = *(const float4*)(scale + c);
  float4 sh = *(const float4*)(shift + c);
  v.x = fmaxf(fmaf(v.x, sc.x, sh.x), 0.f);
  v.y = fmaxf(fmaf(v.y, sc.y, sh.y), 0.f);
  v.z = fmaxf(fmaf(v.z, sc.z, sh.z), 0.f);
  v.w = fmaxf(fmaf(v.w, sc.w, sh.w), 0.f);
  *(float4*)(data + i * 4) = v;
}

// BN1 stats over 800k incidences of msg_pre = P[nidx] + Q[eidx].
// Wave per incidence row (float4/lane = 128 cols), 16 incidences per wave,
// register accumulation -> LDS -> one global atomic per column per block.
#define ST_PER_WAVE 16
__global__ __launch_bounds__(256) void inc_stats(
    const float* __restrict__ P, const float* __restrict__ Q,
    const int* __restrict__ nidx, const int* __restrict__ eidx,
    float* __restrict__ sumG, float* __restrict__ sqG) {
  __shared__ float s_sum[128];
  __shared__ float s_sq[128];
  const int tid = threadIdx.x, lane = tid & 31, wv = tid >> 5;
  if (tid < 128) {
    s_sum[tid] = 0.f;
    s_sq[tid] = 0.f;
  }
  __syncthreads();
  const int base = (blockIdx.x * 8 + wv) * ST_PER_WAVE;
  const int c = lane << 2;
  float4 s = make_float4(0, 0, 0, 0), s2 = make_float4(0, 0, 0, 0);
  for (int j = 0; j < ST_PER_WAVE; ++j) {
    const int i = base + j;
    const long long n = nidx[i], e = eidx[i];
    float4 p = *(const float4*)(P + n * Hdim + c);
    float4 q = *(const float4*)(Q + e * Hdim + c);
    const float x0 = p.x + q.x, x1 = p.y + q.y, x2 = p.z + q.z, x3 = p.w + q.w;
    s.x += x0; s2.x += x0 * x0;
    s.y += x1; s2.y += x1 * x1;
    s.z += x2; s2.z += x2 * x2;
    s.w += x3; s2.w += x3 * x3;
  }
  atomicAdd(&s_sum[c + 0], s.x); atomicAdd(&s_sq[c + 0], s2.x);
  atomicAdd(&s_sum[c + 1], s.y); atomicAdd(&s_sq[c + 1], s2.y);
  atomicAdd(&s_sum[c + 2], s.z); atomicAdd(&s_sq[c + 2], s2.z);
  atomicAdd(&s_sum[c + 3], s.w); atomicAdd(&s_sq[c + 3], s2.w);
  __syncthreads();
  if (tid < 128)
    unsafeAtomicAdd(&sumG[tid], s_sum[tid]);
  else
    unsafeAtomicAdd(&sqG[tid - 128], s_sq[tid - 128]);
}

// cat[e, :128] += msg ; cat[e, 128:] += node_rep[n]
#define SC_PER_WAVE 4
__global__ __launch_bounds__(256) void inc_scatter(
    const float* __restrict__ P, const float* __restrict__ Q,
    const float* __restrict__ node_rep, const int* __restrict__ nidx,
    const int* __restrict__ eidx, const float* __restrict__ scale,
    const float* __restrict__ shift, float* __restrict__ cat) {
  const int tid = threadIdx.x, lane = tid & 31, wv = tid >> 5;
  const int base = (blockIdx.x * 8 + wv) * SC_PER_WAVE;
  const int c = lane << 2;
  const float4 sc = *(const float4*)(scale + c);
  const float4 sh = *(const float4*)(shift + c);
  for (int j = 0; j < SC_PER_WAVE; ++j) {
    const int i = base + j;
    const long long n = nidx[i], e = eidx[i];
    float4 p = *(const float4*)(P + n * Hdim + c);
    float4 q = *(const float4*)(Q + e * Hdim + c);
    float4 nr = *(const float4*)(node_rep + n * Hdim + c);
    float* cl = cat + e * (2 * Hdim) + c;
    unsafeAtomicAdd(cl + 0, fmaxf(fmaf(p.x + q.x, sc.x, sh.x), 0.f));
    unsafeAtomicAdd(cl + 1, fmaxf(fmaf(p.y + q.y, sc.y, sh.y), 0.f));
    unsafeAtomicAdd(cl + 2, fmaxf(fmaf(p.z + q.z, sc.z, sh.z), 0.f));
    unsafeAtomicAdd(cl + 3, fmaxf(fmaf(p.w + q.w, sc.w, sh.w), 0.f));
    float* ch = cl + Hdim;
    unsafeAtomicAdd(ch + 0, nr.x);
    unsafeAtomicAdd(ch + 1, nr.y);
    unsafeAtomicAdd(ch + 2, nr.z);
    unsafeAtomicAdd(ch + 3, nr.w);
  }
}

// lvl_aggr[n] += cat[e, :128] - msg   (msg recomputed from cached P/Q)
__global__ __launch_bounds__(256) void inc_lvl(
    const float* __restrict__ P, const float* __restrict__ Q,
    const float* __restrict__ cat, const int* __restrict__ nidx,
    const int* __restrict__ eidx, const float* __restrict__ scale,
    const float* __restrict__ shift, float* __restrict__ lvl) {
  const int tid = threadIdx.x, lane = tid & 31, wv = tid >> 5;
  const int base = (blockIdx.x * 8 + wv) * SC_PER_WAVE;
  const int c = lane << 2;
  const float4 sc = *(const float4*)(scale + c);
  const float4 sh = *(const float4*)(shift + c);
  for (int j = 0; j < SC_PER_WAVE; ++j) {
    const int i = base + j;
    const long long n = nidx[i], e = eidx[i];
    float4 p = *(const float4*)(P + n * Hdim + c);
    float4 q = *(const float4*)(Q + e * Hdim + c);
    float4 cl = *(const float4*)(cat + e * (2 * Hdim) + c);
    float* lp = lvl + n * Hdim + c;
    unsafeAtomicAdd(lp + 0, cl.x - fmaxf(fmaf(p.x + q.x, sc.x, sh.x), 0.f));
    unsafeAtomicAdd(lp + 1, cl.y - fmaxf(fmaf(p.y + q.y, sc.y, sh.y), 0.f));
    unsafeAtomicAdd(lp + 2, cl.z - fmaxf(fmaf(p.z + q.z, sc.z, sh.z), 0.f));
    unsafeAtomicAdd(lp + 3, cl.w - fmaxf(fmaf(p.w + q.w, sc.w, sh.w), 0.f));
  }
}

extern "C" void kernel_launch(void* const* d_in, const int* in_sizes, int n_in,
                              void* d_out, int out_size, void* d_ws,
                              size_t ws_size, hipStream_t stream) {
  const float* node_rep = (const float*)d_in[0];
  const float* edge_rep = (const float*)d_in[1];
  const int* nidx = (const int*)d_in[2];
  const int* eidx = (const int*)d_in[3];
  const float* W1 = (const float*)d_in[4];
  const float* g1 = (const float*)d_in[5];
  const float* b1 = (const float*)d_in[6];
  const float* W2a = (const float*)d_in[7];
  const float* g2a = (const float*)d_in[8];
  const float* b2a = (const float*)d_in[9];
  const float* W2b = (const float*)d_in[10];
  const float* g2b = (const float*)d_in[11];
  const float* b2b = (const float*)d_in[12];
  const float* W3a = (const float*)d_in[13];
  const float* g3a = (const float*)d_in[14];
  const float* b3a = (const float*)d_in[15];
  const float* W3b = (const float*)d_in[16];
  const float* g3b = (const float*)d_in[17];
  const float* b3b = (const float*)d_in[18];
  const float* eps1 = (const float*)d_in[19];
  const float* eps2 = (const float*)d_in[20];

  float* node_out = (float*)d_out;                      // 100k x 128
  float* edge_out = node_out + (size_t)NN * Hdim;       // 400k x 128

  float* ws = (float*)d_ws;
  float* P = ws;                                // 100k x 128 (later: z1_node)
  float* Q = P + (size_t)NN * Hdim;             // 400k x 128 (later: z1_edge)
  float* cat = Q + (size_t)NE * Hdim;           // 400k x 256
  float* lvl = cat + (size_t)NE * 2 * Hdim;     // 100k x 128
  float* st = lvl + (size_t)NN * Hdim;          // 5 BNs x [sum,sq,scale,shift]
  float* bn1 = st, *bn2a = st + 512, *bn2b = st + 1024, *bn3a = st + 1536,
        *bn3b = st + 2048;

  hipMemsetAsync(cat, 0, (size_t)NE * 2 * Hdim * sizeof(float), stream);
  hipMemsetAsync(lvl, 0, (size_t)NN * Hdim * sizeof(float), stream);
  hipMemsetAsync(st, 0, 5 * 512 * sizeof(float), stream);

  const int gN = (NN + 63) / 64;  // 1563
  const int gE = NE / 64;         // 6250

  // P = node_rep @ W1[:128], Q = edge_rep @ W1[128:]
  gemm128<0, 0><<<gN, 256, 0, stream>>>(node_rep, Hdim, nullptr, 0, nullptr,
                                        nullptr, nullptr, W1, P, NN, nullptr,
                                        nullptr);
  gemm128<0, 0><<<gE, 256, 0, stream>>>(edge_rep, Hdim, nullptr, 0, nullptr,
                                        nullptr, nullptr, W1 + 128 * Hdim, Q,
                                        NE, nullptr, nullptr);
  // BN1 stats + fold
  inc_stats<<<NI / (8 * ST_PER_WAVE), 256, 0, stream>>>(P, Q, nidx, eidx,
                                                        bn1, bn1 + 128);
  bn_finalize<<<1, 128, 0, stream>>>(bn1, bn1 + 128, g1, b1, 1.0f / NI,
                                     bn1 + 256, bn1 + 384);
  // cat segment sums, then lvl_aggr
  inc_scatter<<<NI / (8 * SC_PER_WAVE), 256, 0, stream>>>(
      P, Q, node_rep, nidx, eidx, bn1 + 256, bn1 + 384, cat);
  inc_lvl<<<NI / (8 * SC_PER_WAVE), 256, 0, stream>>>(
      P, Q, cat, nidx, eidx, bn1 + 256, bn1 + 384, lvl);

  // Node tower: x = (1+eps1)*node_rep + lvl ; z1 -> BN -> z2 -> BN
  gemm128<1, 1><<<gN, 256, 0, stream>>>(node_rep, Hdim, lvl, Hdim, eps1,
                                        nullptr, nullptr, W2a, P, NN, bn2a,
                                        bn2a + 128);
  bn_finalize<<<1, 128, 0, stream>>>(bn2a, bn2a + 128, g2a, b2a, 1.0f / NN,
                                     bn2a + 256, bn2a + 384);
  gemm128<2, 1><<<gN, 256, 0, stream>>>(P, Hdim, nullptr, 0, nullptr,
                                        bn2a + 256, bn2a + 384, W2b, node_out,
                                        NN, bn2b, bn2b + 128);
  bn_finalize<<<1, 128, 0, stream>>>(bn2b, bn2b + 128, g2b, b2b, 1.0f / NN,
                                     bn2b + 256, bn2b + 384);
  bn_apply<<<(NN * 32 + 255) / 256, 256, 0, stream>>>(node_out, bn2b + 256,
                                                      bn2b + 384,
                                                      (long long)NN * 32);

  // Edge tower: y = (1+eps2)*edge_rep + cat[:,128:]
  gemm128<1, 1><<<gE, 256, 0, stream>>>(edge_rep, Hdim, cat + Hdim, 2 * Hdim,
                                        eps2, nullptr, nullptr, W3a, Q, NE,
                                        bn3a, bn3a + 128);
  bn_finalize<<<1, 128, 0, stream>>>(bn3a, bn3a + 128, g3a, b3a, 1.0f / NE,
                                     bn3a + 256, bn3a + 384);
  gemm128<2, 1><<<gE, 256, 0, stream>>>(Q, Hdim, nullptr, 0, nullptr,
                                        bn3a + 256, bn3a + 384, W3b, edge_out,
                                        NE, bn3b, bn3b + 128);
  bn_finalize<<<1, 128, 0, stream>>>(bn3b, bn3b + 128, g3b, b3b, 1.0f / NE,
                                     bn3b + 256, bn3b + 384);
  bn_apply<<<(NE * 32 + 255) / 256, 256, 0, stream>>>(edge_out, bn3b + 256,
                                                      bn3b + 384,
                                                      (long long)NE * 32);
}